// NonOverlappingConv1d_24885040513344
// MI455X (gfx1250) — compile-verified
//
#include <hip/hip_runtime.h>

typedef __attribute__((ext_vector_type(2))) float v2f;
typedef __attribute__((ext_vector_type(8))) float v8f;

#define CIN     64
#define DLEN    65536
#define OUT_C   64
#define OUT_DIM 16384
#define NT_PER_B 1024          // OUT_DIM / 16 N-tiles per batch image

// Grid: 4096 blocks x 256 threads (8 wave32).
// Wave w: m-tile = w&3 (rows m0..m0+15 of the 64x256 weight matrix),
//         handles two adjacent 16-wide N-tiles (two accumulators for ILP).
// Block covers 4 consecutive N-tile positions x all 4 m-tiles.
__global__ __launch_bounds__(256)
void conv1d_wmma_f32(const float* __restrict__ x,
                     const float* __restrict__ wgt,
                     const float* __restrict__ bias,
                     float* __restrict__ out) {
    const int tid  = threadIdx.x;
    const int wave = tid >> 5;
    const int lane = tid & 31;
    const int lo   = lane & 15;   // column / M-row selector within half-wave
    const int hi   = lane >> 4;   // 0 or 1: selects K/M upper half per ISA layout

    const int m0    = (wave & 3) * 16;
    const int ntpos = blockIdx.x * 4 + (wave >> 2) * 2;   // even; pair stays in one batch
    const int b0    = ntpos / NT_PER_B;
    const int n0    = (ntpos % NT_PER_B) * 16;

    // A-tile (16x4 f32): lane holds W[m0+lo][4*kk + 2*hi + {0,1}]  (VGPR0/1 = K lsb)
    const float* aptr = wgt + (size_t)(m0 + lo) * 256 + 2 * hi;

    // B-tile (4x16 f32) for step kk is the contiguous 64-float run x[b, kk, 4*n0 ..):
    // lane holds B[K = vgpr + 2*hi][N = n0+lo] = x[b, kk, 4*(n0+lo) + 2*hi + vgpr]
    const float* bptr0 = x + (size_t)b0 * ((size_t)CIN * DLEN)
                           + (size_t)(4 * n0 + 4 * lo + 2 * hi);
    const float* bptr1 = bptr0 + 64;   // next N-tile: +16 cols -> +64 floats

    v8f acc0 = {0.f, 0.f, 0.f, 0.f, 0.f, 0.f, 0.f, 0.f};
    v8f acc1 = {0.f, 0.f, 0.f, 0.f, 0.f, 0.f, 0.f, 0.f};

    // K = 256 reduction = 64 steps of V_WMMA_F32_16X16X4_F32 (c index == kk)
    #pragma unroll 4
    for (int kk = 0; kk < 64; ++kk) {
        v2f a  = *(const v2f*)(aptr  + 4 * kk);
        v2f bA = *(const v2f*)(bptr0 + (size_t)kk * DLEN);
        v2f bB = *(const v2f*)(bptr1 + (size_t)kk * DLEN);
        // (neg_a, A, neg_b, B, c_mod, C, reuse_a, reuse_b)
        acc0 = __builtin_amdgcn_wmma_f32_16x16x4_f32(false, a, false, bA,
                                                     (short)0, acc0, false, false);
        acc1 = __builtin_amdgcn_wmma_f32_16x16x4_f32(false, a, false, bB,
                                                     (short)0, acc1, false, false);
    }

    // Epilogue: y = acc/sqrt(64) + 0.1*bias ; C/D layout: VGPR v -> M = v + 8*hi
    const float scale = 0.125f;
    float*       obase = out  + (size_t)b0 * OUT_C * OUT_DIM + (size_t)(n0 + lo);
    const float* bbase = bias + (size_t)(n0 + lo);
    #pragma unroll
    for (int v = 0; v < 8; ++v) {
        const int o = m0 + v + 8 * hi;
        const size_t roff = (size_t)o * OUT_DIM;
        obase[roff]      = acc0[v] * scale + bbase[roff]      * 0.1f;
        obase[roff + 16] = acc1[v] * scale + bbase[roff + 16] * 0.1f;
    }
}

extern "C" void kernel_launch(void* const* d_in, const int* in_sizes, int n_in,
                              void* d_out, int out_size, void* d_ws, size_t ws_size,
                              hipStream_t stream) {
    const float* x    = (const float*)d_in[0];   // [16, 64, 65536] f32
    const float* wgt  = (const float*)d_in[1];   // [64, 64, 1, 4]  f32 (flat 64x256)
    const float* bias = (const float*)d_in[2];   // [1, 64, 16384]  f32
    float* out = (float*)d_out;                  // [16, 64, 16384] f32

    (void)in_sizes; (void)n_in; (void)out_size; (void)d_ws; (void)ws_size;

    dim3 grid(4096);   // 4096 * 4 N-tile positions = 16384 = 16 batches * 1024
    dim3 block(256);   // 8 wave32: 4 m-tiles x 2 N-slot groups
    conv1d_wmma_f32<<<grid, block, 0, stream>>>(x, wgt, bias, out);
}